// ThermostatNN_5085241279188
// MI455X (gfx1250) — compile-verified
//
#include <hip/hip_runtime.h>
#include <math.h>

// ThermostatNN on MI455X (gfx1250, wave32).
//
// Mapping: each wave simulates 16 batch elements. Per step, layer 1 is a
// single-K WMMA chain:  Ht(64x16) = W1t(64x4, K=2 used) x InpT(4x16) + b1
// using 4x V_WMMA_F32_16X16X4_F32 (one per 16-hidden tile).
//   A (loop-invariant): lanes 0-15 = W1^T rows (VGPR0=K0 temp-weight,
//       VGPR1=K1 aux-weight); lanes 16-31 = padded K=2,3 rows = 0.
//   B (per step):       lanes 0-15 = {temp, aux} of 16 elements, lanes 16-31 = 0.
//   C                 = b1 pre-broadcast in C-layout (bias fused into the MMA).
// D holds Ht[j][b] (j = 16t + v + 8*half, b = lane%16). Layer 2 is
// 32x single-instruction ReLU via v_med3_f32(x, 0, 3e38) -- the FINITE upper
// bound stops LLVM folding the med3 back into canonicalize+maxnum -- plus
// packed v_pk_fma_f32 against W2 pre-broadcast in the same layout, split over
// 4 accumulators to break the FMA chain; one __shfl_xor(16) folds the two
// lane-halves into the full 64-wide dot. b2 is folded into the accumulator
// seeds (0.5*b2 per half). Both halves mirror per-element state so control
// flow stays wave-uniform; only lanes 0-15 write the trajectory.
//
// Roofline: ~4 GFLOP vs ~44 MB traffic -> compute (VALU/matrix) bound; the
// WMMA path removes the 128-FMA layer-1 inner loop from the VALU stream
// (f32 WMMAs co-issue with VALU on CDNA5), leaving ~65 VALU/step/16-elems.

typedef float v2f __attribute__((ext_vector_type(2)));
typedef float v8f __attribute__((ext_vector_type(8)));

#define HID     64
#define NSTEP   40
#define TON     66.0f
#define TOFF    78.0f
#define RELU_HI 3.0e38f       // finite: keeps v_med3_f32 from folding to max
#define ELEMS_PER_BLOCK 128   // 8 waves * 16 elements

__global__ __launch_bounds__(256) void thermostat_wmma_kernel(
    const float* __restrict__ x_init,   // [B,4] = {step, isOn, temp, aux}
    const float* __restrict__ W1,       // [2,64]
    const float* __restrict__ b1,       // [64]
    const float* __restrict__ W2,       // [64,1]
    const float* __restrict__ b2,       // [1]
    float* __restrict__ out,            // [40,B]
    int B)
{
    __shared__ float sW1[2 * HID];
    __shared__ float sB1[HID];
    __shared__ float sW2[HID];
    __shared__ float sB2;

    const int tid = threadIdx.x;
    if (tid < 2 * HID)                 sW1[tid] = W1[tid];
    if (tid < HID)                     sB1[tid] = b1[tid];
    if (tid >= HID && tid < 2 * HID)   sW2[tid - HID] = W2[tid - HID];
    if (tid == 2 * HID)                sB2 = b2[0];
    __syncthreads();

    const int  lane = tid & 31;
    const int  wave = tid >> 5;
    const int  col  = lane & 15;          // batch slot within the wave
    const bool hi   = lane >= 16;         // upper half-wave
    const int  elem = blockIdx.x * ELEMS_PER_BLOCK + wave * 16 + col;

    // Loop-invariant WMMA operands.
    v2f at[4];                 // A tiles: W1^T, 16 hidden rows x K(=2 used)
    v8f b1c[4];                // C init: b1 broadcast in C-layout
    v8f w2c[4];                // W2 broadcast in C-layout (for layer-2 dot)
#pragma unroll
    for (int t = 0; t < 4; ++t) {
        at[t].x = hi ? 0.0f : sW1[0 * HID + 16 * t + col];  // K=0: temp weight
        at[t].y = hi ? 0.0f : sW1[1 * HID + 16 * t + col];  // K=1: aux weight
#pragma unroll
        for (int v = 0; v < 8; ++v) {
            const int j = 16 * t + v + (hi ? 8 : 0);        // hidden index
            b1c[t][v] = sB1[j];
            w2c[t][v] = sW2[j];
        }
    }
    const float b2half = 0.5f * sB2;     // seeds one accumulator per half;
                                         // halves are summed -> adds full b2

    // Per-element state (mirrored across both lane-halves): one b128 load.
    const float4 xi = ((const float4*)x_init)[elem];
    float isOn = xi.y;
    float temp = xi.z;
    const float aux  = xi.w;
    const float auxm = hi ? 0.0f : aux;   // B operand K=3 row is padded zero

    for (int s = 0; s < NSTEP; ++s) {
        // B operand: K=0 row = temp, K=1 row = aux, K=2,3 rows = 0.
        v2f bb;
        bb.x = hi ? 0.0f : temp;
        bb.y = auxm;

        // Layer 1 via WMMA, layer-2 partial dot in-register.
        // 4 independent accumulators break the FMA RAW chain (8 deep each).
        float acc[4] = { b2half, 0.0f, 0.0f, 0.0f };
#pragma unroll
        for (int t = 0; t < 4; ++t) {
            v8f d = __builtin_amdgcn_wmma_f32_16x16x4_f32(
                /*neg_a=*/false, at[t], /*neg_b=*/false, bb,
                /*c_mod=*/(short)0, b1c[t],
                /*reuse_a=*/false, /*reuse_b=*/false);
#pragma unroll
            for (int v = 0; v < 8; ++v) {
                // ReLU in ONE instruction: v_med3_f32(x, 0, 3e38).
                const float h = __builtin_amdgcn_fmed3f(d[v], 0.0f, RELU_HI);
                acc[t] = fmaf(h, w2c[t][v], acc[t]);
            }
        }
        const float accs = (acc[0] + acc[1]) + (acc[2] + acc[3]);
        // Each half holds 32 of the 64 hidden contributions (+0.5*b2); fold.
        const float z = accs + __shfl_xor(accs, 16, 32);

        // sigmoid -> plant -> dtemp  (plant = s*10 - 5, dtemp = plant*10)
        const float sig   = __builtin_amdgcn_rcpf(1.0f + __expf(-z));
        const float dtemp = sig * 100.0f - 50.0f;

        // Bang-bang dynamics (reference's `active` mask is always true).
        const bool  off = (isOn <= 0.5f);
        const float tn  = temp + dtemp + (off ? 0.0f : 5.0f);
        isOn = off ? (tn <= TON  ? 1.0f : isOn)
                   : (tn <= TOFF ? isOn : 0.0f);
        temp = tn;

        if (!hi) out[s * B + elem] = temp;   // traj[s][elem]
    }
}

extern "C" void kernel_launch(void* const* d_in, const int* in_sizes, int n_in,
                              void* d_out, int out_size, void* d_ws, size_t ws_size,
                              hipStream_t stream) {
    const float* x_init = (const float*)d_in[0];
    const float* W1     = (const float*)d_in[1];
    const float* b1     = (const float*)d_in[2];
    const float* W2     = (const float*)d_in[3];
    const float* b2     = (const float*)d_in[4];
    float* out          = (float*)d_out;

    const int B = in_sizes[0] / 4;                  // x_init is [B,4]
    const int grid = (B + ELEMS_PER_BLOCK - 1) / ELEMS_PER_BLOCK;

    thermostat_wmma_kernel<<<grid, 256, 0, stream>>>(x_init, W1, b1, W2, b2, out, B);
}